// SAGE_31894427140507
// MI455X (gfx1250) — compile-verified
//
#include <hip/hip_runtime.h>

// ---------------------------------------------------------------------------
// GraphSAGE (mean aggregator), 3 layers, inference.
//   h_neigh = segment_mean(h[src], dst);  out = h@W_self + h_neigh@W_neigh + b
// Layers 0,1: F=128->128 + ReLU.  Layer 2: 128->47.
//
// MI455X reasoning:
//  * h (100k x 128 f32 = 51.2MB) fits in 192MB L2 -> edge gather/scatter is an
//    L2-resident atomic problem; one wave per edge, float4 per lane (coalesced
//    512B rows), f32 atomicAdd resolves in L2.
//  * GEMM (~20 GFLOP total) is trivial vs. aggregation -> keep full fp32 with
//    V_WMMA_F32_16X16X4_F32, fusing bias + mean-normalize + ReLU.
//  * Weights (64KB) are pre-transposed once per call so BOTH fragment loads
//    (A from h/agg, B from Wt) are contiguous global_load_b64 with immediate
//    offsets; N_OUT/RELU are template params so the k-loop has no address ALU.
// ---------------------------------------------------------------------------

#define F_DIM 128

typedef __attribute__((ext_vector_type(2))) float v2f;
typedef __attribute__((ext_vector_type(8))) float v8f;

static __device__ __forceinline__ v8f wmma_f32x4(v2f a, v2f b, v8f c) {
  // 8 args: (neg_a, A, neg_b, B, c_mod, C, reuse_a, reuse_b)
  return __builtin_amdgcn_wmma_f32_16x16x4_f32(
      false, a, false, b, (short)0, c, false, false);
}

// ---------------------------------------------------------------------------
__global__ void __launch_bounds__(256) zero_f32(float* __restrict__ p, size_t n) {
  size_t i = (size_t)blockIdx.x * blockDim.x + threadIdx.x;
  size_t stride = (size_t)gridDim.x * blockDim.x;
  for (; i < n; i += stride) p[i] = 0.0f;
}

__global__ void __launch_bounds__(256) count_deg(const int* __restrict__ dst,
                                                 float* __restrict__ deg, int nE) {
  int e = blockIdx.x * blockDim.x + threadIdx.x;
  if (e < nE) atomicAdd(&deg[dst[e]], 1.0f);
}

__global__ void __launch_bounds__(256) invert_deg(float* __restrict__ deg, int n) {
  int i = blockIdx.x * blockDim.x + threadIdx.x;
  if (i < n) deg[i] = 1.0f / fmaxf(deg[i], 1.0f);   // matches max(deg,1)
}

// W[k][n] (K=128 x n_out, row-major) -> WT[n][k] (n_out x 128)
__global__ void __launch_bounds__(256)
transpose_w(const float* __restrict__ W, float* __restrict__ WT, int n_out) {
  int i = blockIdx.x * blockDim.x + threadIdx.x;
  if (i >= F_DIM * n_out) return;
  int k = i / n_out;
  int n = i - k * n_out;
  WT[n * F_DIM + k] = W[i];
}

// One wave (32 lanes) per edge; each lane moves one float4 (128 floats/row).
__global__ void __launch_bounds__(256)
edge_scatter(const float* __restrict__ h, const int* __restrict__ src,
             const int* __restrict__ dst, float* __restrict__ agg, int nE) {
  size_t t = (size_t)blockIdx.x * blockDim.x + threadIdx.x;
  int e = (int)(t >> 5);
  int lane = (int)(t & 31);
  if (e >= nE) return;
  int s = src[e];
  int d = dst[e];
  const float4* __restrict__ hp = (const float4*)(h + (size_t)s * F_DIM);
  float4 v = hp[lane];
  float* ap = agg + (size_t)d * F_DIM + lane * 4;
  atomicAdd(ap + 0, v.x);
  atomicAdd(ap + 1, v.y);
  atomicAdd(ap + 2, v.z);
  atomicAdd(ap + 3, v.w);
}

// Fused SAGE layer GEMM: out = h@Ws + (agg*inv_deg)@Wn + b, optional ReLU.
// One wave computes one 16x16 output tile with V_WMMA_F32_16X16X4_F32.
//
// Fragment layouts (wave32, ISA 7.12.2):
//   A 16x4 f32 (2 VGPRs): lanes 0-15 -> M=lane, K={k,k+1};
//                         lanes 16-31 -> M=lane-16, K={k+2,k+3}
//   B 4x16 f32 (2 VGPRs): mirrored (lane&15 selects N, lane>>4 selects K pair)
//   C/D (8 VGPRs): VGPR j -> row j (lanes 0-15) / row j+8 (lanes 16-31), N=lane&15
template <int N_OUT, bool RELU>
__global__ void __launch_bounds__(256)
sage_gemm(const float* __restrict__ h, const float* __restrict__ agg,
          const float* __restrict__ inv_deg,
          const float* __restrict__ WsT, const float* __restrict__ WnT,
          const float* __restrict__ bias, float* __restrict__ out, int n_nodes) {
  constexpr int TILES_N = (N_OUT + 15) / 16;
  constexpr bool PARTIAL = (N_OUT % 16) != 0;

  int wave = blockIdx.x * (blockDim.x >> 5) + (threadIdx.x >> 5);
  int tiles_m = (n_nodes + 15) >> 4;
  if (wave >= tiles_m * TILES_N) return;          // wave-uniform exit

  int tm = wave / TILES_N;
  int tn = wave - tm * TILES_N;
  int lane = threadIdx.x & 31;
  int half = lane >> 4;                            // 0 or 1
  int l16 = lane & 15;
  int koff = half * 2;                             // K-pair base for this lane

  int mrow = tm * 16 + l16;
  if (mrow >= n_nodes) mrow = n_nodes - 1;         // arithmetic clamp (EXEC stays full)
  int ncol = tn * 16 + l16;
  int ncl = ncol;
  float nmask = 1.0f;
  if constexpr (PARTIAL) {
    ncl = (ncol < N_OUT) ? ncol : (N_OUT - 1);
    nmask = (ncol < N_OUT) ? 1.0f : 0.0f;
  }

  float invd = inv_deg[mrow];
  // float2 bases with the lane's K-pair offset folded in: all k-loop loads
  // become immediate-offset global_load_b64.
  const float2* __restrict__ hp = (const float2*)(h + (size_t)mrow * F_DIM) + (koff >> 1);
  const float2* __restrict__ ap = (const float2*)(agg + (size_t)mrow * F_DIM) + (koff >> 1);
  const float2* __restrict__ wsp = (const float2*)(WsT + (size_t)ncl * F_DIM) + (koff >> 1);
  const float2* __restrict__ wnp = (const float2*)(WnT + (size_t)ncl * F_DIM) + (koff >> 1);

  v8f c;
  float bv = bias[ncl];
  if constexpr (PARTIAL) bv *= nmask;
#pragma unroll
  for (int j = 0; j < 8; ++j) c[j] = bv;

  // Self term: h @ W_self
#pragma unroll 8
  for (int k = 0; k < F_DIM; k += 4) {
    float2 av = hp[k >> 1];
    float2 bw = wsp[k >> 1];
    v2f a{av.x, av.y};
    v2f b{bw.x, bw.y};
    if constexpr (PARTIAL) { b.x *= nmask; b.y *= nmask; }
    c = wmma_f32x4(a, b, c);
  }
  // Neighbor term: (agg * inv_deg) @ W_neigh  (mean fold-in on the A fragment)
#pragma unroll 8
  for (int k = 0; k < F_DIM; k += 4) {
    float2 av = ap[k >> 1];
    float2 bw = wnp[k >> 1];
    v2f a{av.x * invd, av.y * invd};
    v2f b{bw.x, bw.y};
    if constexpr (PARTIAL) { b.x *= nmask; b.y *= nmask; }
    c = wmma_f32x4(a, b, c);
  }

  if constexpr (RELU) {
#pragma unroll
    for (int j = 0; j < 8; ++j) c[j] = fmaxf(c[j], 0.0f);
  }

#pragma unroll
  for (int j = 0; j < 8; ++j) {
    int row = tm * 16 + j + half * 8;
    bool ok = (row < n_nodes);
    if constexpr (PARTIAL) ok = ok && (ncol < N_OUT);
    if (ok) out[(size_t)row * N_OUT + ncol] = c[j];
  }
}

// ---------------------------------------------------------------------------
extern "C" void kernel_launch(void* const* d_in, const int* in_sizes, int n_in,
                              void* d_out, int out_size, void* d_ws, size_t ws_size,
                              hipStream_t stream) {
  const float* x   = (const float*)d_in[0];
  const int*   src = (const int*)d_in[1];
  const int*   dst = (const int*)d_in[2];
  const float* Ws[3] = { (const float*)d_in[3], (const float*)d_in[6], (const float*)d_in[9]  };
  const float* Wn[3] = { (const float*)d_in[4], (const float*)d_in[7], (const float*)d_in[10] };
  const float* bs[3] = { (const float*)d_in[5], (const float*)d_in[8], (const float*)d_in[11] };

  const int N = in_sizes[0] / F_DIM;   // 100000
  const int E = in_sizes[1];           // 1600000
  const int NC = 47;

  // Workspace: inv_deg[N] | agg[N*128] | hA[N*128] | hB[N*128] | 6x W^T
  float* invdeg = (float*)d_ws;
  float* agg = invdeg + (size_t)N;
  float* hA  = agg + (size_t)N * F_DIM;
  float* hB  = hA  + (size_t)N * F_DIM;
  float* wt  = hB  + (size_t)N * F_DIM;
  float* WsT[3] = { wt,                 wt + 2 * F_DIM * F_DIM, wt + 4 * F_DIM * F_DIM };
  float* WnT[3] = { wt + F_DIM * F_DIM, wt + 3 * F_DIM * F_DIM, wt + 4 * F_DIM * F_DIM + F_DIM * NC };

  const int ZB = 2048;                               // grid-stride zero blocks

  // Degrees (graph shared by all layers; once per call)
  zero_f32<<<ZB, 256, 0, stream>>>(invdeg, (size_t)N);
  count_deg<<<(E + 255) / 256, 256, 0, stream>>>(dst, invdeg, E);
  invert_deg<<<(N + 255) / 256, 256, 0, stream>>>(invdeg, N);

  // Pre-transpose weights (tiny: <=64KB each) -> contiguous B-fragment loads
  int nout_l[3] = { F_DIM, F_DIM, NC };
  for (int l = 0; l < 3; ++l) {
    int elems = F_DIM * nout_l[l];
    transpose_w<<<(elems + 255) / 256, 256, 0, stream>>>(Ws[l], WsT[l], nout_l[l]);
    transpose_w<<<(elems + 255) / 256, 256, 0, stream>>>(Wn[l], WnT[l], nout_l[l]);
  }

  const float* hin = x;
  float* houts[3] = { hA, hB, (float*)d_out };

  size_t scatter_threads = (size_t)E * 32;
  int scatter_blocks = (int)((scatter_threads + 255) / 256);
  int tiles_m = (N + 15) / 16;

  for (int l = 0; l < 3; ++l) {
    zero_f32<<<ZB, 256, 0, stream>>>(agg, (size_t)N * F_DIM);
    edge_scatter<<<scatter_blocks, 256, 0, stream>>>(hin, src, dst, agg, E);

    if (l < 2) {
      int waves = tiles_m * ((F_DIM + 15) / 16);
      int blocks = (waves + 7) / 8;                  // 8 waves / 256-thread block
      sage_gemm<F_DIM, true><<<blocks, 256, 0, stream>>>(
          hin, agg, invdeg, WsT[l], WnT[l], bs[l], houts[l], N);
    } else {
      int waves = tiles_m * ((NC + 15) / 16);
      int blocks = (waves + 7) / 8;
      sage_gemm<NC, false><<<blocks, 256, 0, stream>>>(
          hin, agg, invdeg, WsT[l], WnT[l], bs[l], houts[l], N);
    }
    hin = houts[l];
  }
}